// CurveGraphic2d_68685116998042
// MI455X (gfx1250) — compile-verified
//
#include <hip/hip_runtime.h>

typedef __attribute__((ext_vector_type(2))) float v2f;
typedef __attribute__((ext_vector_type(8))) float v8f;

#define NB 64
#define NS 15
#define CANVAS_H 256
#define CANVAS_W 256
#define PIX (CANVAS_H * CANVAS_W)
#define CENTER 128.0f

// ---------------------------------------------------------------------------
// Kernel 1: evaluate Bezier samples and pack the per-batch WMMA A-matrix.
// A is 16x4 (samples x K), K-components = [-2*sy, -2*sx, 1, |s|^2]
// (coords centered at 128). ISA layout for 32-bit A 16x4:
//   lane L holds row M = L&15; VGPR0 = K(0 or 2), VGPR1 = K(1 or 3)
//   (lanes 0-15 -> K0,K1 ; lanes 16-31 -> K2,K3)
// One thread per (batch, lane): 64*32 = 2048 threads.
// Also precomputes per-batch shading params {1/width, aa} so the hot loop
// never divides.
// ---------------------------------------------------------------------------
__global__ void bezier_setup_kernel(const float* __restrict__ inp,
                                    const float* __restrict__ widths,
                                    const float* __restrict__ aaf,
                                    float* __restrict__ a0mat,
                                    float* __restrict__ a1mat,
                                    float* __restrict__ params) {
    int tid  = blockIdx.x * blockDim.x + threadIdx.x;   // 0..2047
    int b    = tid >> 5;
    int lane = tid & 31;
    int m    = lane & 15;
    int s    = (m < NS) ? m : (NS - 1);                 // row 15 dups sample 14

    float t  = (float)s * (1.0f / (float)(NS - 1));
    float u  = 1.0f - t;
    float w0 = u * u * u;
    float w1 = 3.0f * u * u * t;
    float w2 = 3.0f * u * t * t;
    float w3 = t * t * t;

    const float* p = inp + b * 8;                       // [4,2] = (y,x) pairs
    float sy = 256.0f * (w0 * p[0] + w1 * p[2] + w2 * p[4] + w3 * p[6]) - CENTER;
    float sx = 256.0f * (w0 * p[1] + w1 * p[3] + w2 * p[5] + w3 * p[7]) - CENTER;

    float a0, a1;
    if (lane < 16) { a0 = -2.0f * sy;  a1 = -2.0f * sx; }          // K0, K1
    else           { a0 = 1.0f;        a1 = sy * sy + sx * sx; }   // K2, K3

    a0mat[tid] = a0;
    a1mat[tid] = a1;

    if (lane == 0) {
        params[2 * b + 0] = 1.0f / widths[b];   // full-precision divide, once
        params[2 * b + 1] = aaf[b];
    }
}

// ---------------------------------------------------------------------------
// Kernel 2: per wave, 32 consecutive pixels (two 16-pixel WMMA tiles),
// loop over 64 batches. D[m,n] = d^2(sample m, pixel n) via
// V_WMMA_F32_16X16X4_F32. Min over samples = min over 8 acc VGPRs (v_min3
// tree) + ONE SWAPX16 ds_swizzle: swap tiles between halves first, then a
// single cross-half min leaves tile0's full min in lanes 0-15 and tile1's
// in lanes 16-31 -- exactly where each lane's output pixel lives.
// Epilogue: raw v_sqrt, mul by precomputed 1/width, exp2(aa*log2(x)).
// ---------------------------------------------------------------------------
__global__ void curve_render_kernel(const float* __restrict__ a0mat,
                                    const float* __restrict__ a1mat,
                                    const float* __restrict__ params,
                                    float* __restrict__ out) {
    int tid   = blockIdx.x * blockDim.x + threadIdx.x;  // 0..65535, exact cover
    int lane  = tid & 31;
    int pbase = tid & ~31;                              // wave's 32-pixel base
    int n     = lane & 15;
    bool lo   = (lane < 16);

    // B-matrix (4x16, pixels): K-components [py, px, |p|^2, 1], centered.
    // VGPR v, lane L -> K = v + (L<16 ? 0 : 2), column N = L&15.
    int p0i = pbase + n;
    int p1i = pbase + 16 + n;
    float y0 = (float)(p0i & 255) - CENTER, x0 = (float)(p0i >> 8) - CENTER;
    float y1 = (float)(p1i & 255) - CENTER, x1 = (float)(p1i >> 8) - CENTER;

    v2f B0, B1;
    B0.x = lo ? y0 : (y0 * y0 + x0 * x0);
    B0.y = lo ? x0 : 1.0f;
    B1.x = lo ? y1 : (y1 * y1 + x1 * x1);
    B1.y = lo ? x1 : 1.0f;

    float* outp = out + pbase + lane;

    for (int b = 0; b < NB; ++b) {
        v2f A;
        A.x = a0mat[b * 32 + lane];
        A.y = a1mat[b * 32 + lane];

        v8f c0 = {};
        v8f c1 = {};
        // 8 args: (neg_a, A, neg_b, B, c_mod, C, reuse_a, reuse_b)
        c0 = __builtin_amdgcn_wmma_f32_16x16x4_f32(false, A, false, B0,
                                                   (short)0, c0, false, false);
        c1 = __builtin_amdgcn_wmma_f32_16x16x4_f32(false, A, false, B1,
                                                   (short)0, c1, false, false);

        // min over the 8 sample rows held in this lane-half
        float m0 = fminf(fminf(fminf(c0[0], c0[1]), fminf(c0[2], c0[3])),
                         fminf(fminf(c0[4], c0[5]), fminf(c0[6], c0[7])));
        float m1 = fminf(fminf(fminf(c1[0], c1[1]), fminf(c1[2], c1[3])),
                         fminf(fminf(c1[4], c1[5]), fminf(c1[6], c1[7])));

        // own tile's half-min stays put; other tile's half-min crosses halves
        float v = lo ? m0 : m1;
        float u = lo ? m1 : m0;
        u = __int_as_float(
                __builtin_amdgcn_ds_swizzle(__float_as_int(u), 0x401F)); // SWAPX16
        float d2 = fminf(v, u);   // full 16-sample min for this lane's pixel

        float dist  = __builtin_amdgcn_sqrtf(fmaxf(d2, 0.0f));  // raw v_sqrt_f32
        float inv_w = params[2 * b + 0];
        float aa    = params[2 * b + 1];
        float base  = dist * inv_w + 1e-6f;                     // base >= 1e-6
        float val   = __builtin_amdgcn_exp2f(aa * __builtin_amdgcn_logf(base));
        float r     = 1.0f - val;
        r = fminf(fmaxf(r, 0.0f), 1.0f);

        outp[b * PIX] = r;   // coalesced 128B per wave
    }
}

extern "C" void kernel_launch(void* const* d_in, const int* in_sizes, int n_in,
                              void* d_out, int out_size, void* d_ws, size_t ws_size,
                              hipStream_t stream) {
    const float* inputs = (const float*)d_in[0];   // [64,4,2]
    const float* widths = (const float*)d_in[1];   // [64]
    const float* aaf    = (const float*)d_in[2];   // [64]
    float* out          = (float*)d_out;           // [64,256,256]

    float* a0mat  = (float*)d_ws;                  // 2048 floats
    float* a1mat  = a0mat + NB * 32;               // 2048 floats
    float* params = a1mat + NB * 32;               // 128 floats  (~16.5KB total)

    bezier_setup_kernel<<<8, 256, 0, stream>>>(inputs, widths, aaf,
                                               a0mat, a1mat, params);
    curve_render_kernel<<<PIX / 256, 256, 0, stream>>>(a0mat, a1mat, params, out);
}